// LightConv_69526930588082
// MI455X (gfx1250) — compile-verified
//
#include <hip/hip_runtime.h>
#include <math.h>

// LightConv on MI455X (gfx1250): depthwise conv1d over the flat-reshaped
// (8192 rows x 4096) fp32 buffer, 7-tap softmax kernel per head, zero pad 3.
// Memory-bound (268 MB @ 23.3 TB/s ~= 11.5 us floor) -> stage rows in LDS via
// CDNA5 async global->LDS copies, compute from LDS, 128-bit stores.

#define T_LEN      4096
#define KW         7
#define NH         16
#define THREADS    256
#define PER_THREAD 16                 // T_LEN / THREADS
#define LDS_FLOATS (T_LEN + 8)        // [0..3] left pad, row at [4..4099], [4100..4103] right pad

#if defined(__has_builtin)
# if __has_builtin(__builtin_amdgcn_global_load_async_to_lds_b128)
#  define HAVE_ASYNC_B128 1
# endif
# if __has_builtin(__builtin_amdgcn_s_wait_asynccnt)
#  define HAVE_WAIT_ASYNC 1
# endif
#endif

typedef int v4i __attribute__((ext_vector_type(4)));
typedef __attribute__((address_space(1))) v4i as1_v4i;   // global (__device__) int4
typedef __attribute__((address_space(3))) v4i as3_v4i;   // LDS (__shared__) int4
typedef float v4f __attribute__((ext_vector_type(4)));

__device__ __forceinline__ void copy16_to_lds(const float* g, float* l) {
#ifdef HAVE_ASYNC_B128
  // async DMA: 16 bytes global -> LDS, tracked by ASYNCcnt
  __builtin_amdgcn_global_load_async_to_lds_b128(
      (as1_v4i*)(void*)g, (as3_v4i*)(void*)l, /*offset=*/0, /*cpol=*/0);
#else
  *(v4f*)l = *(const v4f*)g;   // fallback: global_load_b128 + ds_store_b128
#endif
}

__device__ __forceinline__ void wait_async_copies() {
#ifdef HAVE_ASYNC_B128
# ifdef HAVE_WAIT_ASYNC
  __builtin_amdgcn_s_wait_asynccnt(0);
# else
  asm volatile("s_wait_asynccnt 0" ::: "memory");
# endif
#endif
}

__global__ __launch_bounds__(THREADS)
void lightconv_kernel(const float* __restrict__ x,
                      const float* __restrict__ weight,
                      float* __restrict__ out) {
  __shared__ float lds[LDS_FLOATS];

  const int row = blockIdx.x;            // row = n*16 + h, 8192 rows total
  const int tid = threadIdx.x;
  const float* src = x + (size_t)row * T_LEN;
  float*       dst = out + (size_t)row * T_LEN;

  // Zero the halo cells (indices 0..3 and 4100..4103).
  if (tid < 4)       lds[tid] = 0.0f;
  else if (tid < 8)  lds[T_LEN + tid] = 0.0f;

  // Kick off async global->LDS copies: 1024 chunks of 16B, 4 per lane.
#pragma unroll
  for (int it = 0; it < 4; ++it) {
    const int chunk = tid + it * THREADS;          // 0..1023
    copy16_to_lds(src + chunk * 4, &lds[4 + chunk * 4]);
  }

  // Softmax of this head's 7 taps (overlaps with the DMA).
  const int h = row & (NH - 1);
  float wv[KW];
  float m = -1e30f;
#pragma unroll
  for (int k = 0; k < KW; ++k) {
    wv[k] = weight[h * KW + k];
    m = fmaxf(m, wv[k]);
  }
  float ssum = 0.0f;
#pragma unroll
  for (int k = 0; k < KW; ++k) {
    wv[k] = __expf(wv[k] - m);
    ssum += wv[k];
  }
  const float inv = 1.0f / ssum;
#pragma unroll
  for (int k = 0; k < KW; ++k) wv[k] *= inv;

  wait_async_copies();
  __syncthreads();

  // Each thread: 24 contiguous LDS floats (aligned b128 loads), 16 outputs.
  const int base = tid * PER_THREAD;               // lds[base + i] = x[s = base + i - 4]
  float xv[PER_THREAD + 8];
#pragma unroll
  for (int i = 0; i < (PER_THREAD + 8) / 4; ++i)
    *(v4f*)&xv[i * 4] = *(const v4f*)&lds[base + i * 4];

  float acc[PER_THREAD];
#pragma unroll
  for (int j = 0; j < PER_THREAD; ++j) {
    float a = 0.0f;
#pragma unroll
    for (int k = 0; k < KW; ++k)
      a = fmaf(wv[k], xv[j + k + 1], a);           // x[s + k - 3] -> xv[j+k+1]
    acc[j] = a;
  }

#pragma unroll
  for (int i = 0; i < PER_THREAD / 4; ++i) {
    v4f o = { acc[i * 4 + 0], acc[i * 4 + 1], acc[i * 4 + 2], acc[i * 4 + 3] };
    *(v4f*)&dst[base + i * 4] = o;                 // global_store_b128
  }
}

extern "C" void kernel_launch(void* const* d_in, const int* in_sizes, int n_in,
                              void* d_out, int out_size, void* d_ws, size_t ws_size,
                              hipStream_t stream) {
  (void)n_in; (void)out_size; (void)d_ws; (void)ws_size;
  const float* x = (const float*)d_in[0];
  const float* w = (const float*)d_in[1];
  float* out = (float*)d_out;
  const int n_rows = in_sizes[0] / T_LEN;          // 8192
  lightconv_kernel<<<dim3(n_rows), dim3(THREADS), 0, stream>>>(x, w, out);
}